// ChebyConv_72645076845146
// MI455X (gfx1250) — compile-verified
//
#include <hip/hip_runtime.h>

typedef __attribute__((ext_vector_type(2))) float v2f;
typedef __attribute__((ext_vector_type(8))) float v8f;

// ---------------------------------------------------------------------------
// Kernel 1: rows[] is sorted -> build CSR row pointers (N+1 entries).
// ---------------------------------------------------------------------------
__global__ void build_rowptr(const int* __restrict__ rows, int* __restrict__ row_ptr,
                             int nE, int nN) {
    int e = blockIdx.x * blockDim.x + threadIdx.x;
    if (e >= nE) return;
    int r = rows[e];
    int rp = (e == 0) ? -1 : rows[e - 1];
    for (int rr = rp + 1; rr <= r; ++rr) row_ptr[rr] = e;
    if (e == nE - 1) {
        for (int rr = r + 1; rr <= nN; ++rr) row_ptr[rr] = nE;
    }
}

// ---------------------------------------------------------------------------
// Kernel 2/3: SpMM, one wave32 per output node. lane l owns features l, l+32
// -> each gathered row x[col] is two coalesced 128B loads. vals/cols indices
// are wave-uniform -> scalar loads. out = scale*acc - sub (sub may be null).
// ---------------------------------------------------------------------------
__global__ void spmm_kernel(const float* __restrict__ h, const int* __restrict__ row_ptr,
                            const int* __restrict__ cols, const float* __restrict__ vals,
                            float* __restrict__ out, const float* __restrict__ sub,
                            float scale, int nN) {
    int wave = (int)((blockIdx.x * blockDim.x + threadIdx.x) >> 5);
    int lane = threadIdx.x & 31;
    if (wave >= nN) return;
    int start = row_ptr[wave];
    int end   = row_ptr[wave + 1];
    float acc0 = 0.0f, acc1 = 0.0f;
    for (int e = start; e < end; ++e) {
        float v = vals[e];
        int   c = cols[e];
        const float* xr = h + (size_t)c * 64;
        if (e + 8 < end) {   // prefetch the gather row a few edges ahead
            int cn = cols[e + 8];
            __builtin_prefetch(h + (size_t)cn * 64 + lane, 0, 0);
        }
        acc0 = fmaf(v, xr[lane], acc0);
        acc1 = fmaf(v, xr[lane + 32], acc1);
    }
    size_t o = (size_t)wave * 64;
    if (sub != nullptr) {
        out[o + lane]      = scale * acc0 - sub[o + lane];
        out[o + lane + 32] = scale * acc1 - sub[o + lane + 32];
    } else {
        out[o + lane]      = acc0;
        out[o + lane + 32] = acc1;
    }
}

// ---------------------------------------------------------------------------
// Kernel 4: out = Tx0@W0 + Tx1@W1 + Tx2@W2 + bias via V_WMMA_F32_16X16X4_F32.
// Block = 256 threads = 8 waves; each wave computes a 16x64 output stripe
// (4 x v8f accumulators). Weights staged in LDS *swizzled* so that each B
// fragment (W[k][n], W[k+1][n]) is a contiguous float2:
//     w_lds[t*4096 + (k>>1)*128 + n*2 + (k&1)] = W[t][k][n]
// -> every B operand is one aligned ds_load_b64 straight into a VGPR pair
//    (no v_mov repacking in the WMMA inner loop).
//
// fp32 WMMA layouts (ISA 7.12.2):
//   A 16x4:  lanes 0-15 -> M=lane, K = kk+{0,1}; lanes 16-31 -> M=lane-16, K = kk+{2,3}
//   B 4x16:  mirrored (lane half selects K pair, lane&15 selects N)
//   C 16x16: VGPR j -> M = j + 8*(lane>=16), N = lane&15
// ---------------------------------------------------------------------------
__device__ __forceinline__ void gemm_accum_64(const float* __restrict__ A,
                                              const float* __restrict__ Wl,
                                              int ksel, int bn, v8f acc[4]) {
#pragma unroll
    for (int kk = 0; kk < 64; kk += 4) {
        // A fragment: two consecutive f32 at 8B-aligned offset -> global_load_b64
        v2f a = *(const v2f*)(A + kk + ksel);
        int kprow = ((kk + ksel) >> 1) * 128;   // swizzled k-pair row in LDS
#pragma unroll
        for (int nt = 0; nt < 4; ++nt) {
            // B fragment: contiguous float2 in swizzled LDS -> ds_load_b64
            v2f b = *(const v2f*)(Wl + kprow + (nt * 16 + bn) * 2);
            acc[nt] = __builtin_amdgcn_wmma_f32_16x16x4_f32(
                false, a, false, b, (short)0, acc[nt], false, false);
        }
    }
}

__global__ void cheb_gemm(const float* __restrict__ Tx0, const float* __restrict__ Tx1,
                          const float* __restrict__ Tx2, const float* __restrict__ W,
                          const float* __restrict__ bias, float* __restrict__ out, int nN) {
    __shared__ float w_lds[3 * 64 * 64];
    // cooperative swizzled stage: coalesced global reads, scattered LDS writes
    for (int i = threadIdx.x; i < 3 * 64 * 64; i += blockDim.x) {
        int t = i >> 12;            // /4096
        int k = (i >> 6) & 63;
        int n = i & 63;
        w_lds[(t << 12) + ((k >> 1) << 7) + (n << 1) + (k & 1)] = W[i];
    }
    __syncthreads();

    int wave = threadIdx.x >> 5;
    int lane = threadIdx.x & 31;
    int m0 = (blockIdx.x * 8 + wave) * 16;

    int arow = m0 + (lane & 15);
    int arow_c = (arow < nN) ? arow : (nN - 1);   // clamp tail loads (writes masked)
    int ksel = (lane >> 4) << 1;                  // 0 for lanes 0-15, 2 for 16-31
    int bn = lane & 15;

    v8f acc[4] = {};
    gemm_accum_64(Tx0 + (size_t)arow_c * 64, w_lds,            ksel, bn, acc);
    gemm_accum_64(Tx1 + (size_t)arow_c * 64, w_lds + (1 << 12), ksel, bn, acc);
    gemm_accum_64(Tx2 + (size_t)arow_c * 64, w_lds + (2 << 12), ksel, bn, acc);

    // store with fused bias; C VGPR j -> row m0 + 8*(lane>=16) + j, col nt*16 + (lane&15)
    int mbase = m0 + ((lane >> 4) << 3);
#pragma unroll
    for (int nt = 0; nt < 4; ++nt) {
        int col = nt * 16 + bn;
        float bv = bias[col];
#pragma unroll
        for (int j = 0; j < 8; ++j) {
            int row = mbase + j;
            if (row < nN) out[(size_t)row * 64 + col] = acc[nt][j] + bv;
        }
    }
}

// ---------------------------------------------------------------------------
// Launch: Tx1 = A x ; Tx2 = 2 A Tx1 - x ; out = [Tx0 Tx1 Tx2] @ [W0;W1;W2] + b
// ---------------------------------------------------------------------------
extern "C" void kernel_launch(void* const* d_in, const int* in_sizes, int n_in,
                              void* d_out, int out_size, void* d_ws, size_t ws_size,
                              hipStream_t stream) {
    const float* x    = (const float*)d_in[0];
    const int*   rows = (const int*)d_in[1];
    const int*   cols = (const int*)d_in[2];
    const float* vals = (const float*)d_in[3];
    const float* W    = (const float*)d_in[4];
    const float* bias = (const float*)d_in[5];

    const int N = in_sizes[0] / 64;   // F_IN = 64
    const int E = in_sizes[1];

    char* ws = (char*)d_ws;
    int* row_ptr = (int*)ws;
    size_t off = (((size_t)(N + 1) * sizeof(int)) + 255) & ~(size_t)255;
    float* Tx1 = (float*)(ws + off);
    float* Tx2 = Tx1 + (size_t)N * 64;

    build_rowptr<<<(E + 255) / 256, 256, 0, stream>>>(rows, row_ptr, E, N);

    int spmm_grid = (N + 7) / 8;       // 8 waves (nodes) per 256-thread block
    spmm_kernel<<<spmm_grid, 256, 0, stream>>>(x,   row_ptr, cols, vals, Tx1, nullptr, 1.0f, N);
    spmm_kernel<<<spmm_grid, 256, 0, stream>>>(Tx1, row_ptr, cols, vals, Tx2, x,       2.0f, N);

    cheb_gemm<<<(N + 127) / 128, 256, 0, stream>>>(x, Tx1, Tx2, W, bias, (float*)d_out, N);
}